// MixingAttention_91319594648320
// MI455X (gfx1250) — compile-verified
//
#include <hip/hip_runtime.h>
#include <math.h>

#define N_TOK 64
#define DIM   192
#define CA    96
#define HEADS 6
#define HD    16
#define NWIN  2048
#define BIMG  32

typedef __attribute__((ext_vector_type(2))) float v2f;
typedef __attribute__((ext_vector_type(8))) float v8f;
typedef unsigned int u32x4 __attribute__((ext_vector_type(4)));
typedef int i32x4 __attribute__((ext_vector_type(4)));
typedef int i32x8 __attribute__((ext_vector_type(8)));

#define BNS 0.9999950000374997f   // 1/sqrt(1+1e-5)

#if defined(__has_builtin)
#if __has_builtin(__builtin_amdgcn_tensor_load_to_lds) && __has_builtin(__builtin_amdgcn_s_wait_tensorcnt)
#define HAVE_TDM 1
#else
#define HAVE_TDM 0
#endif
#else
#define HAVE_TDM 0
#endif

__device__ __forceinline__ float gelu_f(float x) {
  return 0.5f * x * (1.0f + erff(x * 0.7071067811865475f));
}
__device__ __forceinline__ float sigm_f(float x) {
  return 1.0f / (1.0f + __expf(-x));
}

__device__ __forceinline__ v8f wmma4(v2f a, v2f b, v8f c) {
  // D = A(16x4,f32) * B(4x16,f32) + C(16x16,f32)
  return __builtin_amdgcn_wmma_f32_16x16x4_f32(false, a, false, b, (short)0, c, false, false);
}
__device__ __forceinline__ v8f zero8() {
  v8f z = {0.f,0.f,0.f,0.f,0.f,0.f,0.f,0.f};
  return z;
}
// A fragment: row-major A[M][K], lanes 0-15 -> M rows; k pair (k0,k0+1) low half, (k0+2,k0+3) high half
__device__ __forceinline__ v2f afrag(const float* A, int ld, int row0, int k0, int lane) {
  int r  = row0 + (lane & 15);
  int kk = k0 + ((lane >> 4) << 1);
  v2f v; v.x = A[r * ld + kk]; v.y = A[r * ld + kk + 1];
  return v;
}
// B fragment: row-major B[K][N]
__device__ __forceinline__ v2f bfrag(const float* B, int ld, int k0, int col0, int lane) {
  int c  = col0 + (lane & 15);
  int kk = k0 + ((lane >> 4) << 1);
  v2f v; v.x = B[kk * ld + c]; v.y = B[(kk + 1) * ld + c];
  return v;
}
// B fragment from transposed storage: B[k][n] = S[n*ld + k]  (used for Q @ K^T)
__device__ __forceinline__ v2f bfragT(const float* S, int ld, int k0, int col0, int lane) {
  int c  = col0 + (lane & 15);
  int kk = k0 + ((lane >> 4) << 1);
  v2f v; v.x = S[c * ld + kk]; v.y = S[c * ld + kk + 1];
  return v;
}

// LDS byte offset of a generic pointer into __shared__ (flat LDS aperture: low 32 bits)
__device__ __forceinline__ unsigned lds_addr_of(const void* p) {
  return (unsigned)(size_t)p;
}

#if HAVE_TDM
// TDM: DMA a (tile_w x tile_h) f32 tile (row stride = stride_elems) from global into
// contiguous LDS at lds_byte_off.  D# packing per CDNA5 ISA 8.3/8.4 (2-D tensor,
// groups 2/3 zero).  Issue from ONE wave only; EXEC is ignored by TDM.
__device__ __forceinline__ void tdm_load_f32(unsigned lds_byte_off, const float* gsrc,
                                             unsigned tensor_w, unsigned tensor_h,
                                             unsigned tile_w, unsigned tile_h,
                                             unsigned stride_elems) {
  unsigned long long ga = (unsigned long long)(size_t)gsrc;
  u32x4 g0;
  g0.x = 1u;                                        // count=1 (valid user descriptor)
  g0.y = lds_byte_off;                              // lds_addr
  g0.z = (unsigned)ga;                              // global_addr[31:0]
  g0.w = (unsigned)((ga >> 32) & 0x1FFFFFFull) | 0x80000000u;  // addr[56:32] | type=2
  i32x8 g1;
  g1[0] = (int)(2u << 16);                          // data_size = 4 bytes
  g1[1] = (int)((tensor_w & 0xFFFFu) << 16);        // tensor_dim0[15:0]
  g1[2] = (int)((tensor_w >> 16) | ((tensor_h & 0xFFFFu) << 16));
  g1[3] = (int)((tensor_h >> 16) | (tile_w << 16)); // tile_dim0
  g1[4] = (int)tile_h;                              // tile_dim1 (tile_dim2 = 0)
  g1[5] = (int)stride_elems;                        // tensor_dim0_stride[31:0]
  g1[6] = 0;
  g1[7] = 0;
  i32x4 z4 = {0, 0, 0, 0};
#if __clang_major__ >= 23
  i32x8 z8 = {0, 0, 0, 0, 0, 0, 0, 0};
  __builtin_amdgcn_tensor_load_to_lds(g0, g1, z4, z4, z8, 0);
#else
  __builtin_amdgcn_tensor_load_to_lds(g0, g1, z4, z4, 0);
#endif
  __builtin_amdgcn_s_wait_tensorcnt(0);
}
#endif

// ---------------------------------------------------------------------------
// K1: per-window  Ya = LN(x@w_pa + b_pa) -> xat ;  Yc = LN(x@w_pc + b_pc) -> NCHW image
// ---------------------------------------------------------------------------
__global__ __launch_bounds__(256) void k1_proj_ln(
    const float* __restrict__ x,
    const float* __restrict__ w_pa, const float* __restrict__ b_pa,
    const float* __restrict__ ln_a_g, const float* __restrict__ ln_a_b,
    const float* __restrict__ w_pc, const float* __restrict__ b_pc,
    const float* __restrict__ ln_c_g, const float* __restrict__ ln_c_b,
    float* __restrict__ xat, float* __restrict__ xcimg)
{
  __shared__ float sY[N_TOK * DIM];                 // 48 KB (reused A then C)
  const int win = blockIdx.x;
  const int b = win >> 6, hw = (win >> 3) & 7, ww = win & 7;
  const int tid = threadIdx.x, lane = tid & 31, wv = tid >> 5;
  const float* xw = x + (size_t)win * N_TOK * DIM;

  // ---- phase A: 64x96 = 64x192 @ 192x96 ----
  for (int t = wv; t < 24; t += 8) {
    int mt = t & 3, nt = t >> 2;
    v8f acc = zero8();
    for (int k = 0; k < 48; ++k) {
      v2f a  = afrag(xw,   DIM, mt * 16, k * 4, lane);
      v2f bb = bfrag(w_pa, CA,  k * 4,  nt * 16, lane);
      acc = wmma4(a, bb, acc);
    }
    int c = nt * 16 + (lane & 15);
    int rbase = mt * 16 + ((lane >> 4) << 3);
    float bias = b_pa[c];
#pragma unroll
    for (int r = 0; r < 8; ++r) sY[(rbase + r) * CA + c] = acc[r] + bias;
  }
  __syncthreads();
  if (tid < N_TOK) {
    float m = 0.f, v = 0.f;
    for (int c = 0; c < CA; ++c) m += sY[tid * CA + c];
    m *= (1.0f / CA);
    for (int c = 0; c < CA; ++c) { float d = sY[tid * CA + c] - m; v += d * d; }
    float inv = rsqrtf(v * (1.0f / CA) + 1e-5f);
    float* o = xat + (size_t)win * N_TOK * CA + tid * CA;
    for (int c = 0; c < CA; ++c)
      o[c] = (sY[tid * CA + c] - m) * inv * ln_a_g[c] + ln_a_b[c];
  }
  __syncthreads();

  // ---- phase C: 64x192 = 64x192 @ 192x192 ----
  for (int t = wv; t < 48; t += 8) {
    int mt = t & 3, nt = t >> 2;
    v8f acc = zero8();
    for (int k = 0; k < 48; ++k) {
      v2f a  = afrag(xw,   DIM, mt * 16, k * 4, lane);
      v2f bb = bfrag(w_pc, DIM, k * 4,  nt * 16, lane);
      acc = wmma4(a, bb, acc);
    }
    int c = nt * 16 + (lane & 15);
    int rbase = mt * 16 + ((lane >> 4) << 3);
    float bias = b_pc[c];
#pragma unroll
    for (int r = 0; r < 8; ++r) sY[(rbase + r) * DIM + c] = acc[r] + bias;
  }
  __syncthreads();
  if (tid < N_TOK) {
    float m = 0.f, v = 0.f;
    for (int c = 0; c < DIM; ++c) m += sY[tid * DIM + c];
    m *= (1.0f / DIM);
    for (int c = 0; c < DIM; ++c) { float d = sY[tid * DIM + c] - m; v += d * d; }
    float inv = rsqrtf(v * (1.0f / DIM) + 1e-5f);
    int y = hw * 8 + (tid >> 3), xx = ww * 8 + (tid & 7);
    for (int c = 0; c < DIM; ++c) {
      float val = (sY[tid * DIM + c] - m) * inv * ln_c_g[c] + ln_c_b[c];
      xcimg[(((size_t)b * DIM + c) * 64 + y) * 64 + xx] = val;
    }
  }
}

// ---------------------------------------------------------------------------
// K2: depthwise 3x3 SAME + bias + BN + GELU, fused HxW mean  (one block per (b,ch))
//     input plane staged to LDS by the Tensor Data Mover
// ---------------------------------------------------------------------------
__global__ __launch_bounds__(256) void k2_dwconv(
    const float* __restrict__ xcimg, const float* __restrict__ dw_w,
    const float* __restrict__ dw_b, const float* __restrict__ bn1_g,
    const float* __restrict__ bn1_b, float* __restrict__ xc1,
    float* __restrict__ pooled)
{
  __shared__ float sP[4096];                        // 16 KB plane
  __shared__ float red[256];
  const int bc = blockIdx.x;
  const int ch = bc % DIM;
  const float* img = xcimg + (size_t)bc * 4096;
  float* out = xc1 + (size_t)bc * 4096;
  const int tid = threadIdx.x;
#if HAVE_TDM
  if (tid < 32) tdm_load_f32(lds_addr_of(sP), img, 4096, 1, 4096, 1, 4096);
#else
  for (int i = tid; i < 4096; i += 256) sP[i] = img[i];
#endif
  float w[9];
#pragma unroll
  for (int i = 0; i < 9; ++i) w[i] = dw_w[ch * 9 + i];
  const float bias = dw_b[ch], g = bn1_g[ch] * BNS, bb = bn1_b[ch];
  __syncthreads();
  float lsum = 0.f;
  for (int i = 0; i < 16; ++i) {
    int pix = tid + i * 256;
    int y = pix >> 6, xx = pix & 63;
    float s = 0.f;
#pragma unroll
    for (int ky = 0; ky < 3; ++ky) {
      int yy = y + ky - 1;
      if ((unsigned)yy < 64u) {
#pragma unroll
        for (int kx = 0; kx < 3; ++kx) {
          int x2 = xx + kx - 1;
          if ((unsigned)x2 < 64u) s += w[ky * 3 + kx] * sP[yy * 64 + x2];
        }
      }
    }
    float v = gelu_f((s + bias) * g + bb);
    out[pix] = v;
    lsum += v;
  }
  red[tid] = lsum;
  __syncthreads();
  for (int s = 128; s > 0; s >>= 1) {
    if (tid < s) red[tid] += red[tid + s];
    __syncthreads();
  }
  if (tid == 0) pooled[bc] = red[0] * (1.0f / 4096.0f);
}

// ---------------------------------------------------------------------------
// K4: channel-interaction MLP 192 -> 24 -> 96, sigmoid -> gate
// ---------------------------------------------------------------------------
__global__ __launch_bounds__(128) void k4_ci(
    const float* __restrict__ pooled, const float* __restrict__ ci_w1,
    const float* __restrict__ ci_b1, const float* __restrict__ bn_ci_g,
    const float* __restrict__ bn_ci_b, const float* __restrict__ ci_w2,
    const float* __restrict__ ci_b2, float* __restrict__ gate)
{
  __shared__ float sH[24];
  const int b = blockIdx.x, t = threadIdx.x;
  if (t < 24) {
    float a = ci_b1[t];
    for (int c = 0; c < DIM; ++c) a += ci_w1[t * DIM + c] * pooled[b * DIM + c];
    sH[t] = gelu_f(a * BNS * bn_ci_g[t] + bn_ci_b[t]);
  }
  __syncthreads();
  if (t < CA) {
    float a = ci_b2[t];
    for (int h = 0; h < 24; ++h) a += ci_w2[t * 24 + h] * sH[h];
    gate[b * CA + t] = sigm_f(a);
  }
}

// ---------------------------------------------------------------------------
// K5: 1x1 conv 192 -> 96 as WMMA GEMM over 64-pixel slabs
//     B slab (192 x 64, row stride 4096) staged to LDS by TDM 2-D descriptor
// ---------------------------------------------------------------------------
__global__ __launch_bounds__(256) void k5_prconv(
    const float* __restrict__ xc1, const float* __restrict__ pr_w,
    const float* __restrict__ pr_b, float* __restrict__ xc2)
{
  __shared__ float sB[DIM * 64];                    // 48 KB
  const int img = blockIdx.x >> 6, slab = blockIdx.x & 63;
  const int pix0 = slab * 64;
  const int tid = threadIdx.x, lane = tid & 31, wv = tid >> 5;
#if HAVE_TDM
  if (wv == 0)
    tdm_load_f32(lds_addr_of(sB), xc1 + (size_t)img * DIM * 4096 + pix0,
                 4096, DIM, 64, DIM, 4096);
#else
  for (int i = tid; i < DIM * 64; i += 256) {
    int k = i >> 6, j = i & 63;
    sB[i] = xc1[((size_t)(img * DIM + k)) * 4096 + pix0 + j];
  }
#endif
  __syncthreads();
  for (int t = wv; t < 24; t += 8) {               // 6 mt x 4 nt tiles
    int mt = t % 6, nt = t / 6;
    v8f acc = zero8();
    for (int k = 0; k < 48; ++k) {
      v2f a  = afrag(pr_w, DIM, mt * 16, k * 4, lane);
      v2f bb = bfrag(sB,   64,  k * 4,  nt * 16, lane);
      acc = wmma4(a, bb, acc);
    }
    int colp = nt * 16 + (lane & 15);
    int obase = mt * 16 + ((lane >> 4) << 3);
#pragma unroll
    for (int r = 0; r < 8; ++r) {
      int o = obase + r;
      xc2[((size_t)(img * CA + o)) * 4096 + pix0 + colp] = acc[r] + pr_b[o];
    }
  }
}

// ---------------------------------------------------------------------------
// K6a: windowed attention, one head per round (qkv slice GEMM, QK^T, softmax, PV)
//      x_at window staged once to LDS by TDM
// ---------------------------------------------------------------------------
__global__ __launch_bounds__(256) void k6a_attn(
    const float* __restrict__ xat, const float* __restrict__ qkv_w,
    const float* __restrict__ qkv_b, const float* __restrict__ gate,
    const float* __restrict__ rel_tab, float* __restrict__ xat1)
{
  __shared__ float sX[N_TOK * CA];                                  // 24 KB
  __shared__ float sQ[N_TOK * HD], sK[N_TOK * HD], sV[N_TOK * HD];  // 12 KB
  __shared__ float sS[N_TOK * N_TOK];                               // 16 KB
  const int win = blockIdx.x, b = win >> 6;
  const int tid = threadIdx.x, lane = tid & 31, wv = tid >> 5;
  const float* xa = xat + (size_t)win * N_TOK * CA;
#if HAVE_TDM
  if (wv == 0) tdm_load_f32(lds_addr_of(sX), xa, N_TOK * CA, 1, N_TOK * CA, 1, N_TOK * CA);
#else
  for (int i = tid; i < N_TOK * CA; i += 256) sX[i] = xa[i];
#endif
  __syncthreads();

  for (int h = 0; h < HEADS; ++h) {
    // qkv slices for this head: 64 x 48 = 64x96 @ 96x48 (3 parts of 16 cols)
    for (int t = wv; t < 12; t += 8) {
      int part = t >> 2, mt = t & 3;
      int col0 = part * CA + h * HD;
      v8f acc = zero8();
      for (int k = 0; k < 24; ++k) {
        v2f a  = afrag(sX,    CA,     mt * 16, k * 4, lane);
        v2f bb = bfrag(qkv_w, 3 * CA, k * 4,  col0,   lane);
        acc = wmma4(a, bb, acc);
      }
      int d = lane & 15;
      int rbase = mt * 16 + ((lane >> 4) << 3);
      float bias = qkv_b[col0 + d];
      float gm = (part == 2) ? gate[b * CA + h * HD + d] : 1.0f;
      float* dst = (part == 0) ? sQ : (part == 1) ? sK : sV;
#pragma unroll
      for (int r = 0; r < 8; ++r) dst[(rbase + r) * HD + d] = (acc[r] + bias) * gm;
    }
    __syncthreads();
    // S = Q @ K^T   (64x64, K=16)
    for (int t = wv; t < 16; t += 8) {
      int mt = t & 3, nt = t >> 2;
      v8f acc = zero8();
      for (int k = 0; k < 4; ++k) {
        v2f a  = afrag(sQ,  HD, mt * 16, k * 4, lane);
        v2f bb = bfragT(sK, HD, k * 4,  nt * 16, lane);
        acc = wmma4(a, bb, acc);
      }
      int c = nt * 16 + (lane & 15);
      int rbase = mt * 16 + ((lane >> 4) << 3);
#pragma unroll
      for (int r = 0; r < 8; ++r) sS[(rbase + r) * 64 + c] = acc[r];
    }
    __syncthreads();
    // scale by HD^-0.5 and add relative-position bias
    for (int i = tid; i < 4096; i += 256) {
      int n = i >> 6, m = i & 63;
      int idx = ((n >> 3) - (m >> 3) + 7) * 15 + ((n & 7) - (m & 7) + 7);
      sS[i] = sS[i] * 0.25f + rel_tab[idx * HEADS + h];
    }
    __syncthreads();
    // row softmax
    if (tid < N_TOK) {
      float mx = -1e30f;
      for (int m = 0; m < 64; ++m) mx = fmaxf(mx, sS[tid * 64 + m]);
      float sum = 0.f;
      for (int m = 0; m < 64; ++m) {
        float e = __expf(sS[tid * 64 + m] - mx);
        sS[tid * 64 + m] = e;
        sum += e;
      }
      float inv = 1.0f / sum;
      for (int m = 0; m < 64; ++m) sS[tid * 64 + m] *= inv;
    }
    __syncthreads();
    // O = P @ V  (64x16, K=64) -> global
    if (wv < 4) {
      int mt = wv;
      v8f acc = zero8();
      for (int k = 0; k < 16; ++k) {
        v2f a  = afrag(sS, 64, mt * 16, k * 4, lane);
        v2f bb = bfrag(sV, HD, k * 4,  0,      lane);
        acc = wmma4(a, bb, acc);
      }
      int d = lane & 15;
      int rbase = mt * 16 + ((lane >> 4) << 3);
#pragma unroll
      for (int r = 0; r < 8; ++r)
        xat1[(size_t)win * N_TOK * CA + (rbase + r) * CA + h * HD + d] = acc[r];
    }
    __syncthreads();
  }
}

// ---------------------------------------------------------------------------
// K6b: spatial gate + BN + in-place LN + concat, then 192x192 projection
//      attention half staged via TDM into contiguous sZa; conv half in sZc
// ---------------------------------------------------------------------------
__global__ __launch_bounds__(256) void k6b_epi(
    const float* __restrict__ xat1, const float* __restrict__ xc2,
    const float* __restrict__ si_w1, const float* __restrict__ si_b1,
    const float* __restrict__ bn_si_g, const float* __restrict__ bn_si_b,
    const float* __restrict__ si_w2, const float* __restrict__ si_b2,
    const float* __restrict__ cn_g, const float* __restrict__ cn_b,
    const float* __restrict__ an_g, const float* __restrict__ an_b,
    const float* __restrict__ proj_w, const float* __restrict__ proj_b,
    float* __restrict__ out)
{
  __shared__ float sZa[N_TOK * CA];                 // 24 KB (attention half)
  __shared__ float sZc[N_TOK * CA];                 // 24 KB (conv half)
  const int win = blockIdx.x, b = win >> 6, hw = (win >> 3) & 7, ww = win & 7;
  const int tid = threadIdx.x, lane = tid & 31, wv = tid >> 5;
  const float* xa = xat1 + (size_t)win * N_TOK * CA;
#if HAVE_TDM
  if (wv == 0) tdm_load_f32(lds_addr_of(sZa), xa, N_TOK * CA, 1, N_TOK * CA, 1, N_TOK * CA);
#else
  for (int i = tid; i < N_TOK * CA; i += 256) sZa[i] = xa[i];
#endif
  __syncthreads();
  if (tid < N_TOK) {
    const int n = tid;
    // spatial interaction: 96 -> 12 -> 1, sigmoid gate
    float t1[12];
    for (int hh = 0; hh < 12; ++hh) {
      float a = si_b1[hh];
      for (int c = 0; c < CA; ++c) a += si_w1[hh * CA + c] * sZa[n * CA + c];
      t1[hh] = gelu_f(a * BNS * bn_si_g[hh] + bn_si_b[hh]);
    }
    float s = si_b2[0];
    for (int hh = 0; hh < 12; ++hh) s += si_w2[hh] * t1[hh];
    s = sigm_f(s);
    int y = hw * 8 + (n >> 3), xx = ww * 8 + (n & 7);
    int pix = y * 64 + xx;
    for (int c = 0; c < CA; ++c) {
      float v = s * xc2[((size_t)(b * CA + c)) * 4096 + pix];
      sZc[n * CA + c] = v * BNS * cn_g[c] + cn_b[c];
    }
    // LN (an) in place on attention half
    float m = 0.f, vv = 0.f;
    for (int c = 0; c < CA; ++c) m += sZa[n * CA + c];
    m *= (1.0f / CA);
    for (int c = 0; c < CA; ++c) { float d = sZa[n * CA + c] - m; vv += d * d; }
    float inv = rsqrtf(vv * (1.0f / CA) + 1e-5f);
    for (int c = 0; c < CA; ++c)
      sZa[n * CA + c] = (sZa[n * CA + c] - m) * inv * an_g[c] + an_b[c];
  }
  __syncthreads();
  // out = [sZa | sZc] @ proj_w + proj_b    (64x192 @ 192x192)
  for (int t = wv; t < 48; t += 8) {
    int mt = t & 3, nt = t >> 2;
    v8f acc = zero8();
    for (int k = 0; k < 48; ++k) {
      v2f a = (k < 24) ? afrag(sZa, CA, mt * 16, k * 4, lane)
                       : afrag(sZc, CA, mt * 16, (k - 24) * 4, lane);
      v2f bb = bfrag(proj_w, DIM, k * 4, nt * 16, lane);
      acc = wmma4(a, bb, acc);
    }
    int c = nt * 16 + (lane & 15);
    int rbase = mt * 16 + ((lane >> 4) << 3);
    float bias = proj_b[c];
#pragma unroll
    for (int r = 0; r < 8; ++r)
      out[(size_t)win * N_TOK * DIM + (rbase + r) * DIM + c] = acc[r] + bias;
  }
}

// ---------------------------------------------------------------------------
extern "C" void kernel_launch(void* const* d_in, const int* in_sizes, int n_in,
                              void* d_out, int out_size, void* d_ws, size_t ws_size,
                              hipStream_t stream) {
  (void)in_sizes; (void)n_in; (void)out_size; (void)ws_size;
  const float* x       = (const float*)d_in[0];
  const float* w_pa    = (const float*)d_in[3];
  const float* b_pa    = (const float*)d_in[4];
  const float* ln_a_g  = (const float*)d_in[5];
  const float* ln_a_b  = (const float*)d_in[6];
  const float* w_pc    = (const float*)d_in[7];
  const float* b_pc    = (const float*)d_in[8];
  const float* ln_c_g  = (const float*)d_in[9];
  const float* ln_c_b  = (const float*)d_in[10];
  const float* dw_w    = (const float*)d_in[11];
  const float* dw_b    = (const float*)d_in[12];
  const float* bn1_g   = (const float*)d_in[13];
  const float* bn1_b   = (const float*)d_in[14];
  const float* ci_w1   = (const float*)d_in[15];
  const float* ci_b1   = (const float*)d_in[16];
  const float* bn_ci_g = (const float*)d_in[17];
  const float* bn_ci_b = (const float*)d_in[18];
  const float* ci_w2   = (const float*)d_in[19];
  const float* ci_b2   = (const float*)d_in[20];
  const float* pr_w    = (const float*)d_in[21];
  const float* pr_b    = (const float*)d_in[22];
  const float* qkv_w   = (const float*)d_in[23];
  const float* qkv_b   = (const float*)d_in[24];
  const float* si_w1   = (const float*)d_in[25];
  const float* si_b1   = (const float*)d_in[26];
  const float* bn_si_g = (const float*)d_in[27];
  const float* bn_si_b = (const float*)d_in[28];
  const float* si_w2   = (const float*)d_in[29];
  const float* si_b2   = (const float*)d_in[30];
  const float* cn_g    = (const float*)d_in[31];
  const float* cn_b    = (const float*)d_in[32];
  const float* an_g    = (const float*)d_in[33];
  const float* an_b    = (const float*)d_in[34];
  const float* proj_w  = (const float*)d_in[35];
  const float* proj_b  = (const float*)d_in[36];
  const float* rel_tab = (const float*)d_in[37];
  float* outp = (float*)d_out;

  // workspace layout (floats); xc2 reuses the xcimg region after K2 consumed it
  float* ws = (float*)d_ws;
  size_t off = 0;
  float* xat0  = ws + off; off += (size_t)NWIN * N_TOK * CA;     // 12.58M
  float* xcimg = ws + off; off += (size_t)BIMG * DIM * 4096;     // 25.17M
  float* xc1   = ws + off; off += (size_t)BIMG * DIM * 4096;     // 25.17M
  float* xat1  = ws + off; off += (size_t)NWIN * N_TOK * CA;     // 12.58M
  float* pooled = ws + off; off += (size_t)BIMG * DIM;
  float* gate   = ws + off; off += (size_t)BIMG * CA;
  float* xc2 = xcimg;  // safe reuse: xcimg dead after k2

  k1_proj_ln<<<NWIN, 256, 0, stream>>>(x, w_pa, b_pa, ln_a_g, ln_a_b,
                                       w_pc, b_pc, ln_c_g, ln_c_b, xat0, xcimg);
  k2_dwconv<<<BIMG * DIM, 256, 0, stream>>>(xcimg, dw_w, dw_b, bn1_g, bn1_b,
                                            xc1, pooled);
  k4_ci<<<BIMG, 128, 0, stream>>>(pooled, ci_w1, ci_b1, bn_ci_g, bn_ci_b,
                                  ci_w2, ci_b2, gate);
  k5_prconv<<<BIMG * 64, 256, 0, stream>>>(xc1, pr_w, pr_b, xc2);
  k6a_attn<<<NWIN, 256, 0, stream>>>(xat0, qkv_w, qkv_b, gate, rel_tab, xat1);
  k6b_epi<<<NWIN, 256, 0, stream>>>(xat1, xc2, si_w1, si_b1, bn_si_g, bn_si_b,
                                    si_w2, si_b2, cn_g, cn_b, an_g, an_b,
                                    proj_w, proj_b, outp);
}